// BiRGAT_21010980012227
// MI455X (gfx1250) — compile-verified
//
#include <hip/hip_runtime.h>
#include <math.h>

// ---------------- problem constants (from reference) ----------------
#define NPV    8192                 // nodes per view
#define NVIEWS 3
#define NN     (NPV * NVIEWS)       // 24576
#define EDGES  393216
#define RREL   6
#define BBAS   8
#define HEADS  4
#define HO     128                  // heads*out
#define LABELS 4
#define LDSW   68                   // padded LDS row stride in dwords (bank-conflict free)
#define TILE_FLOATS (16 * LDSW)
#define TILE_BYTES  (TILE_FLOATS * 4)

typedef __attribute__((ext_vector_type(2))) float v2f;
typedef __attribute__((ext_vector_type(8))) float v8f;
typedef __attribute__((ext_vector_type(4))) unsigned int v4u;
typedef __attribute__((ext_vector_type(8))) int v8i_t;
typedef __attribute__((ext_vector_type(4))) int v4i_t;

#if __has_builtin(__builtin_amdgcn_tensor_load_to_lds) && \
    __has_builtin(__builtin_amdgcn_s_wait_tensorcnt)
#define HAVE_TDM 1
#else
#define HAVE_TDM 0
#endif

// ---------------- ordered-uint float max helpers --------------------
__device__ __forceinline__ unsigned f2ord(float f) {
  unsigned u = __float_as_uint(f);
  return u ^ ((unsigned)((int)u >> 31) | 0x80000000u);
}
__device__ __forceinline__ float ord2f(unsigned u) {
  unsigned b = (u & 0x80000000u) ? (u ^ 0x80000000u) : ~u;
  return __uint_as_float(b);
}

#if HAVE_TDM
// Issue one TDM load of a 16-row x 64-dword f32 tile (row stride = `stride`
// elements) into LDS at byte offset lds_off.  TDM pad fields insert 4 dwords
// of padding after every 64 dwords stored -> LDS row stride 68 dwords, which
// matches LDSW and kills bank conflicts "for free" in the DMA engine.
__device__ __forceinline__ void tdm_load_tile(unsigned lds_off,
                                              const float* gsrc, int rem_k,
                                              int stride) {
  unsigned long long ga = (unsigned long long)(uintptr_t)gsrc;
  v4u g0;
  g0[0] = 1u;                                        // count=1, user D#
  g0[1] = lds_off;                                   // LDS byte address
  g0[2] = (unsigned)(ga & 0xFFFFFFFFull);            // global addr lo
  g0[3] = (unsigned)((ga >> 32) & 0x1FFFFFFu)        // global addr hi (57b)
          | (2u << 30);                              // type = 2 (image)
  v8i_t g1;
  g1[0] = (2 << 16)        // data_size = 4 bytes
        | (1 << 20)        // pad_enable
        | (5 << 22)        // pad_interval: 64 dwords
        | (3 << 25);       // pad_amount : 4 dwords
  g1[1] = (rem_k & 0xFFFF) << 16;                    // tensor_dim0 lo16
  g1[2] = ((rem_k >> 16) & 0xFFFF) | (16 << 16);     // dim0 hi | tensor_dim1=16
  g1[3] = (64 << 16);                                // dim1 hi=0 | tile_dim0=64
  g1[4] = 16;                                        // tile_dim1=16, tile_dim2=0
  g1[5] = stride;                                    // tensor_dim0_stride lo32
  g1[6] = 0;                                         // stride hi | dim1_stride lo
  g1[7] = 0;
  v4i_t gz4 = {0, 0, 0, 0};                          // 2-D: groups 2/3 unused
  v8i_t gz8 = {0, 0, 0, 0, 0, 0, 0, 0};
  // clang-23 / amdgpu-toolchain 6-arg form (probe-verified arity)
  __builtin_amdgcn_tensor_load_to_lds(g0, g1, gz4, gz4, gz8, 0);
}
#endif

// ---------------- WMMA f32 GEMM: 16-row tile x 128 cols -------------
// Block = 256 threads = 8 wave32 waves; wave w owns cols [16w,16w+16).
// A is [M,K] row-major; TDM (or float4 fallback) stages 16x64 chunks in LDS.
// BT=true : Bm is [128,K] row-major, effective B = Bm^T (pyg Linear W)
// BT=false: Bm is [K,128] row-major (relation weight w[r])
template <bool BT, bool RELU>
__device__ __forceinline__ void gemm_tile_16xK_128(
    const float* __restrict__ A, const float* __restrict__ Bm,
    const float* __restrict__ bias, float* __restrict__ C, int m0, int K,
    float* lds_a) {
  const int wave = threadIdx.x >> 5;
  const int lane = threadIdx.x & 31;
  const int n0 = wave * 16;
  const int rt = lane & 15;  // row (A) / col (B) within 16
  const int kh = lane >> 4;  // 0: K pair {0,1}, 1: K pair {2,3}
  v8f c = {};
#if HAVE_TDM
  const unsigned lds0 = (unsigned)(uintptr_t)lds_a;
  if (wave == 0) tdm_load_tile(lds0, A + (size_t)m0 * K, K, K);
  int buf = 0;
#endif
  for (int kc = 0; kc < K; kc += 64) {
    const int kcw = (K - kc) < 64 ? (K - kc) : 64;  // always multiple of 4 here
#if HAVE_TDM
    // producer wave drains TENSORcnt, barrier publishes buffer to all waves,
    // then the next chunk is DMA'd into the other buffer while we do WMMA.
    if (wave == 0) __builtin_amdgcn_s_wait_tensorcnt(0);
    __syncthreads();
    if (wave == 0 && (kc + 64) < K)
      tdm_load_tile(lds0 + (unsigned)((buf ^ 1) * TILE_BYTES),
                    A + (size_t)m0 * K + kc + 64, K - (kc + 64), K);
    const float* cur = lds_a + buf * TILE_FLOATS;
#else
    {
      const int r = threadIdx.x >> 4;        // 0..15
      const int kk = (threadIdx.x & 15) * 4; // 0..60
      float4 v = make_float4(0.f, 0.f, 0.f, 0.f);
      if (kk < kcw) v = *(const float4*)(A + (size_t)(m0 + r) * K + kc + kk);
      *(float4*)&lds_a[r * LDSW + kk] = v;
    }
    __syncthreads();
    const float* cur = lds_a;
#endif
    for (int k = 0; k < kcw; k += 4) {
      const int kb = k + 2 * kh;
      // A frag (16x4 f32): lanes 0-15 hold K={k,k+1}, lanes 16-31 K={k+2,k+3}
      v2f a = *(const v2f*)&cur[rt * LDSW + kb];
      v2f b;
      if (BT) {
        b = *(const v2f*)(Bm + (size_t)(n0 + rt) * K + kc + kb);
      } else {
        b.x = Bm[(size_t)(kc + kb) * 128 + n0 + rt];
        b.y = Bm[(size_t)(kc + kb + 1) * 128 + n0 + rt];
      }
      c = __builtin_amdgcn_wmma_f32_16x16x4_f32(false, a, false, b, (short)0,
                                                c, false, false);
    }
#if HAVE_TDM
    buf ^= 1;  // next-iteration barrier protects buffer reuse
#else
    __syncthreads();
#endif
  }
  // C/D layout: VGPR r, lanes 0-15 -> M=r, lanes 16-31 -> M=r+8, N=lane%16
  const int col = n0 + rt;
  const float bv = bias ? bias[col] : 0.0f;
  const int mb = m0 + ((lane >> 4) << 3);
#pragma unroll
  for (int r = 0; r < 8; ++r) {
    float v = c[r] + bv;
    if (RELU) v = v > 0.0f ? v : 0.0f;
    C[(size_t)(mb + r) * 128 + col] = v;
  }
}

template <bool RELU>
__global__ __launch_bounds__(256) void proj_kernel(const float* __restrict__ X,
                                                   const float* __restrict__ W,
                                                   const float* __restrict__ b,
                                                   float* __restrict__ H,
                                                   int K) {
  __shared__ float lds_a[2 * TILE_FLOATS];
  gemm_tile_16xK_128<true, RELU>(X, W, b, H, blockIdx.x * 16, K, lds_a);
}

// xw[r] = x @ w[r]   (grid.y = relation)
__global__ __launch_bounds__(256) void xw_kernel(const float* __restrict__ X,
                                                 const float* __restrict__ Wrel,
                                                 float* __restrict__ XW) {
  __shared__ float lds_a[2 * TILE_FLOATS];
  const int r = blockIdx.y;
  gemm_tile_16xK_128<false, false>(X, Wrel + (size_t)r * 128 * 128, nullptr,
                                   XW + (size_t)r * NN * 128, blockIdx.x * 16,
                                   128, lds_a);
}

// w[r] = sum_b comp[r,b] * basis[b]   (R*128*128 outputs)
__global__ __launch_bounds__(256) void relw_kernel(
    const float* __restrict__ comp, const float* __restrict__ basis,
    float* __restrict__ w) {
  int i = blockIdx.x * 256 + threadIdx.x;
  int r = i >> 14;
  int io = i & 16383;
  float acc = 0.0f;
#pragma unroll
  for (int b = 0; b < BBAS; ++b)
    acc += comp[r * BBAS + b] * basis[(size_t)b * 16384 + io];
  w[i] = acc;
}

// si = xw @ q, sj = xw @ k  -- one wave per (relation,node) row.
// Rank-4 q/k means per-edge attention only needs 16B gathers instead of 512B.
__global__ __launch_bounds__(256) void score_kernel(
    const float* __restrict__ xw, const float* __restrict__ q,
    const float* __restrict__ k, float* __restrict__ si,
    float* __restrict__ sj, int RN) {
  int w = (int)((blockIdx.x * 256 + threadIdx.x) >> 5);
  int lane = threadIdx.x & 31;
  if (w >= RN) return;
  float4 xv = ((const float4*)(xw + (size_t)w * 128))[lane];
  float xs[4] = {xv.x, xv.y, xv.z, xv.w};
  float aq[4] = {0, 0, 0, 0}, ak[4] = {0, 0, 0, 0};
#pragma unroll
  for (int j = 0; j < 4; ++j) {
    int o = lane * 4 + j;
#pragma unroll
    for (int h = 0; h < 4; ++h) {
      aq[h] += xs[j] * q[o * 4 + h];
      ak[h] += xs[j] * k[o * 4 + h];
    }
  }
#pragma unroll
  for (int h = 0; h < 4; ++h) {
#pragma unroll
    for (int off = 16; off > 0; off >>= 1) {
      aq[h] += __shfl_down(aq[h], off, 32);
      ak[h] += __shfl_down(ak[h], off, 32);
    }
  }
  if (lane == 0) {
#pragma unroll
    for (int h = 0; h < 4; ++h) {
      si[(size_t)w * 4 + h] = aq[h];
      sj[(size_t)w * 4 + h] = ak[h];
    }
  }
}

// pass 1: alpha = leaky_relu(si[r,dst] + sj[r,src]); segment max; degree
__global__ __launch_bounds__(256) void alpha_kernel(
    const int* __restrict__ src, const int* __restrict__ dst,
    const int* __restrict__ et, const float* __restrict__ si,
    const float* __restrict__ sj, float* __restrict__ alpha,
    unsigned* __restrict__ amax, float* __restrict__ deg) {
  int e = blockIdx.x * 256 + threadIdx.x;
  int r = et[e], s = src[e], d = dst[e];
  size_t gi = ((size_t)r * NN + d) * 4;
  size_t gj = ((size_t)r * NN + s) * 4;
  atomicAdd(&deg[d], 1.0f);
#pragma unroll
  for (int h = 0; h < 4; ++h) {
    float a = si[gi + h] + sj[gj + h];
    a = a > 0.0f ? a : 0.2f * a;
    alpha[(size_t)e * 4 + h] = a;
    atomicMax(&amax[gi + h], f2ord(a));
  }
}

// pass 2: ex = exp(alpha - amax[seg]); segment sum
__global__ __launch_bounds__(256) void exp_kernel(
    const int* __restrict__ dst, const int* __restrict__ et,
    float* __restrict__ alpha, const unsigned* __restrict__ amax,
    float* __restrict__ den) {
  int e = blockIdx.x * 256 + threadIdx.x;
  int r = et[e], d = dst[e];
  size_t gi = ((size_t)r * NN + d) * 4;
#pragma unroll
  for (int h = 0; h < 4; ++h) {
    float am = ord2f(amax[gi + h]);
    float ex = __expf(alpha[(size_t)e * 4 + h] - am);
    alpha[(size_t)e * 4 + h] = ex;
    atomicAdd(&den[gi + h], ex);
  }
}

// pass 3: out[dst] += a * xw[r,src]  (wave per edge; lane -> 4 feats, head=lane/8)
// xw (75.5MB) + out (12.6MB) both fit the 192MB L2 -> gathers/atomics stay on-chip.
__global__ __launch_bounds__(256) void msg_kernel(
    const int* __restrict__ src, const int* __restrict__ dst,
    const int* __restrict__ et, const float* __restrict__ ex,
    const float* __restrict__ den, const float* __restrict__ deg,
    const float* __restrict__ xw, float* __restrict__ out) {
  int e = (int)((blockIdx.x * 256 + threadIdx.x) >> 5);
  int lane = threadIdx.x & 31;
  int r = et[e], s = src[e], d = dst[e];
  int h = lane >> 3;
  size_t gi = ((size_t)r * NN + d) * 4 + h;
  float a = ex[(size_t)e * 4 + h] / (den[gi] + 1e-16f) * deg[d];
  float4 xj = ((const float4*)(xw + ((size_t)r * NN + s) * 128))[lane];
  float* op = out + (size_t)d * 128 + lane * 4;
  atomicAdd(op + 0, a * xj.x);
  atomicAdd(op + 1, a * xj.y);
  atomicAdd(op + 2, a * xj.z);
  atomicAdd(op + 3, a * xj.w);
}

// feat[p] = concat(x[v*NPV+p], v=0..2) @ Wc^T + bc   (thread per (p,label))
__global__ __launch_bounds__(256) void final_kernel(
    const float* __restrict__ x, const float* __restrict__ Wc,
    const float* __restrict__ bc, float* __restrict__ out) {
  int i = blockIdx.x * 256 + threadIdx.x;
  int p = i >> 2;
  int l = i & 3;
  float acc = bc[l];
  for (int v = 0; v < NVIEWS; ++v) {
    const float* row = x + (size_t)(v * NPV + p) * 128;
    const float* wr = Wc + (size_t)l * (NVIEWS * 128) + v * 128;
    for (int j = 0; j < 128; ++j) acc += row[j] * wr[j];
  }
  out[(size_t)p * 4 + l] = acc;
}

extern "C" void kernel_launch(void* const* d_in, const int* in_sizes, int n_in,
                              void* d_out, int out_size, void* d_ws,
                              size_t ws_size, hipStream_t stream) {
  const float* x0 = (const float*)d_in[0];
  const float* x1 = (const float*)d_in[1];
  const float* x2 = (const float*)d_in[2];
  const int* eidx = (const int*)d_in[3];
  const int* etyp = (const int*)d_in[4];
  const float* Wp0 = (const float*)d_in[5];
  const float* bp0 = (const float*)d_in[6];
  const float* Wp1 = (const float*)d_in[7];
  const float* bp1 = (const float*)d_in[8];
  const float* Wp2 = (const float*)d_in[9];
  const float* bp2 = (const float*)d_in[10];
  const float* basis[2] = {(const float*)d_in[11], (const float*)d_in[15]};
  const float* comp[2] = {(const float*)d_in[12], (const float*)d_in[16]};
  const float* qv[2] = {(const float*)d_in[13], (const float*)d_in[17]};
  const float* kv[2] = {(const float*)d_in[14], (const float*)d_in[18]};
  const float* Wc = (const float*)d_in[19];
  const float* bc = (const float*)d_in[20];
  const int* src = eidx;
  const int* dst = eidx + EDGES;

  // workspace layout (~120 MB)
  float* p = (float*)d_ws;
  float* xbuf = p;    p += (size_t)NN * 128;
  float* obuf = p;    p += (size_t)NN * 128;
  float* wrel = p;    p += (size_t)RREL * 128 * 128;
  float* xw = p;      p += (size_t)RREL * NN * 128;
  float* si = p;      p += (size_t)RREL * NN * 4;
  float* sj = p;      p += (size_t)RREL * NN * 4;
  unsigned* amax = (unsigned*)p; p += (size_t)RREL * NN * 4;
  float* den = p;     p += (size_t)RREL * NN * 4;
  float* alpha = p;   p += (size_t)EDGES * 4;
  float* deg = p;     p += (size_t)NN;

  // ---- per-view projections -> x [NN,128] ----
  proj_kernel<true><<<NPV / 16, 256, 0, stream>>>(x0, Wp0, bp0, xbuf, 2000);
  proj_kernel<true><<<NPV / 16, 256, 0, stream>>>(
      x1, Wp1, bp1, xbuf + (size_t)NPV * 128, 1500);
  proj_kernel<true><<<NPV / 16, 256, 0, stream>>>(
      x2, Wp2, bp2, xbuf + (size_t)2 * NPV * 128, 500);

  const float* lin[2] = {xbuf, obuf};
  float* lout[2] = {obuf, xbuf};
  for (int L = 0; L < 2; ++L) {
    relw_kernel<<<(RREL * 128 * 128) / 256, 256, 0, stream>>>(comp[L],
                                                              basis[L], wrel);
    dim3 g(NN / 16, RREL);
    xw_kernel<<<g, 256, 0, stream>>>(lin[L], wrel, xw);
    score_kernel<<<(RREL * NN + 7) / 8, 256, 0, stream>>>(xw, qv[L], kv[L], si,
                                                          sj, RREL * NN);
    (void)hipMemsetAsync(amax, 0, (size_t)RREL * NN * 4 * sizeof(unsigned),
                         stream);
    (void)hipMemsetAsync(den, 0, (size_t)RREL * NN * 4 * sizeof(float), stream);
    (void)hipMemsetAsync(deg, 0, (size_t)NN * sizeof(float), stream);
    (void)hipMemsetAsync(lout[L], 0, (size_t)NN * 128 * sizeof(float), stream);
    alpha_kernel<<<EDGES / 256, 256, 0, stream>>>(src, dst, etyp, si, sj,
                                                  alpha, amax, deg);
    exp_kernel<<<EDGES / 256, 256, 0, stream>>>(dst, etyp, alpha, amax, den);
    msg_kernel<<<EDGES / 8, 256, 0, stream>>>(src, dst, etyp, alpha, den, deg,
                                              xw, lout[L]);
  }
  // layer1 output landed in xbuf
  final_kernel<<<(NPV * LABELS) / 256, 256, 0, stream>>>(xbuf, Wc, bc,
                                                         (float*)d_out);
}